// Rotation_37452114821515
// MI455X (gfx1250) — compile-verified
//
#include <hip/hip_runtime.h>

// 16-byte vector type usable with __builtin_nontemporal_store (clang ext vector).
typedef float v4f __attribute__((ext_vector_type(4)));

// Cayley map for one row: R = ((1-s)I + 2A + 2 v v^T) / (1+s), s = |v|^2.
__device__ __forceinline__ void cayley_row(float x, float y, float z, float* __restrict__ R) {
    float s   = x * x + y * y + z * z;
    float inv = 1.0f / (1.0f + s);      // IEEE divide; memory-bound kernel, accuracy > speed
    float d   = 1.0f - s;
    R[0] = (d + 2.0f * x * x) * inv;
    R[1] = (2.0f * (x * y - z)) * inv;
    R[2] = (2.0f * (x * z + y)) * inv;
    R[3] = (2.0f * (x * y + z)) * inv;
    R[4] = (d + 2.0f * y * y) * inv;
    R[5] = (2.0f * (y * z - x)) * inv;
    R[6] = (2.0f * (x * z - y)) * inv;
    R[7] = (2.0f * (y * z + x)) * inv;
    R[8] = (d + 2.0f * z * z) * inv;
}

// Main kernel: 4 rows per thread.
//   in : 3 x float4  (48 B, 16-B aligned per thread)  -> global_load_b128 x3 (RT: stays in L2)
//   out: 9 x float4  (144 B, 16-B aligned per thread) -> global_store_b128 x9 (NT: don't thrash L2)
__global__ __launch_bounds__(256) void rot_cayley4(const v4f* __restrict__ in4,
                                                   v4f* __restrict__ out4,
                                                   int ngroups) {
    int tid = blockIdx.x * blockDim.x + threadIdx.x;
    if (tid >= ngroups) return;

    const v4f v0 = in4[(size_t)tid * 3 + 0];
    const v4f v1 = in4[(size_t)tid * 3 + 1];
    const v4f v2 = in4[(size_t)tid * 3 + 2];

    float R[36];
    cayley_row(v0.x, v0.y, v0.z, R + 0);   // row 4*tid+0
    cayley_row(v0.w, v1.x, v1.y, R + 9);   // row 4*tid+1
    cayley_row(v1.z, v1.w, v2.x, R + 18);  // row 4*tid+2
    cayley_row(v2.y, v2.z, v2.w, R + 27);  // row 4*tid+3

    v4f* o = out4 + (size_t)tid * 9;
#pragma unroll
    for (int i = 0; i < 9; ++i) {
        v4f w;
        w.x = R[4 * i + 0];
        w.y = R[4 * i + 1];
        w.z = R[4 * i + 2];
        w.w = R[4 * i + 3];
        __builtin_nontemporal_store(w, o + i);  // TH_NT store: write-once output
    }
}

// Scalar tail kernel (rows not covered by groups of 4). For N=4,000,000 this never runs.
__global__ void rot_cayley_tail(const float* __restrict__ t,
                                float* __restrict__ out,
                                int row_start, int nrows) {
    int r = row_start + blockIdx.x * blockDim.x + threadIdx.x;
    if (r >= nrows) return;
    float x = t[(size_t)r * 3 + 0];
    float y = t[(size_t)r * 3 + 1];
    float z = t[(size_t)r * 3 + 2];
    float R[9];
    cayley_row(x, y, z, R);
    float* o = out + (size_t)r * 9;
#pragma unroll
    for (int i = 0; i < 9; ++i) o[i] = R[i];
}

extern "C" void kernel_launch(void* const* d_in, const int* in_sizes, int n_in,
                              void* d_out, int out_size, void* d_ws, size_t ws_size,
                              hipStream_t stream) {
    const float* t = (const float*)d_in[0];
    float* out     = (float*)d_out;

    const long nrows   = (long)in_sizes[0] / 3;   // 4,000,000
    const long ngroups = nrows / 4;               // 1,000,000
    const int  rem     = (int)(nrows % 4);

    if (ngroups > 0) {
        const int threads = 256;
        const long blocks = (ngroups + threads - 1) / threads;  // 3907
        rot_cayley4<<<(unsigned)blocks, threads, 0, stream>>>(
            (const v4f*)t, (v4f*)out, (int)ngroups);
    }
    if (rem > 0) {
        rot_cayley_tail<<<1, 32, 0, stream>>>(t, out, (int)(ngroups * 4), (int)nrows);
    }
}